// DuvenaudMPNN_80470507257955
// MI455X (gfx1250) — compile-verified
//
#include <hip/hip_runtime.h>
#include <hip/hip_bf16.h>

#define N_NODES 20000
#define N_EDGES 640000
#define NV 64
#define NE 16
#define MAXD 8
#define MIND 1
#define T_STEPS 4
#define NOUT 10
#define NGRAPH 32
#define FIN (NV + NE)               // 80
#define DBUCKETS (MAXD - MIND + 1)  // 8

typedef __attribute__((ext_vector_type(2))) float v2f;
typedef __attribute__((ext_vector_type(8))) float v8f;

// ---------------- utility: zero a float region ----------------
__global__ void zero_f32(float* __restrict__ p, int n) {
  int i = blockIdx.x * blockDim.x + threadIdx.x;
  int stride = gridDim.x * blockDim.x;
  for (; i < n; i += stride) p[i] = 0.f;
}

// ---------------- bucket compaction (once per launch) ----------------
__global__ void build_buckets(const int* __restrict__ deg,
                              int* __restrict__ counts,
                              int* __restrict__ lists) {
  int v = blockIdx.x * blockDim.x + threadIdx.x;
  if (v >= N_NODES) return;
  int d = deg[v];
  d = (d < MIND) ? MIND : ((d > MAXD) ? MAXD : d);
  int b = d - MIND;                         // bucket in [0, DBUCKETS)
  int pos = atomicAdd(&counts[b], 1);
  lists[b * N_NODES + pos] = v;
}

// ---------------- edge message scatter: aggr[tgt] += concat(z[src], eattr) ----------------
__global__ void edge_scatter(const float* __restrict__ z,
                             const float* __restrict__ eattr,
                             const int* __restrict__ src,
                             const int* __restrict__ tgt,
                             float* __restrict__ aggr) {
  long long gid = (long long)blockIdx.x * blockDim.x + threadIdx.x;
  if (gid >= (long long)N_EDGES * FIN) return;
  int e = (int)(gid / FIN);
  int f = (int)(gid - (long long)e * FIN);
  float val = (f < NV) ? z[src[e] * NV + f] : eattr[e * NE + (f - NV)];
  atomicAdd(&aggr[tgt[e] * FIN + f], val);
}

// ---------------- degree-bucketed linear + sigmoid via V_WMMA_F32_16X16X4_F32 ----------------
// One wave32 per 16-node tile of one bucket. z_out[v] = sigmoid((aggr[v]/d) @ W_d)
__global__ __launch_bounds__(32)
void bucket_linear_wmma(const float* __restrict__ aggr,
                        const float* __restrict__ Wt,     // layer_weights[t], D*FIN*NV
                        const int* __restrict__ counts,
                        const int* __restrict__ lists,
                        float* __restrict__ zout) {
  const int d = blockIdx.y;                 // bucket
  const int base = blockIdx.x * 16;
  const int count = counts[d];
  if (base >= count) return;                // uniform over the wave -> EXEC stays full

  const int lane = threadIdx.x;             // 0..31
  const float* __restrict__ W = Wt + d * FIN * NV;   // W_d[k*NV + n]
  const float inv_d = 1.0f / (float)(d + MIND);      // 1/clamp(deg,1,8)

  __shared__ float As[16 * 81];             // 16 x 80 A-tile, stride 81 => no bank conflicts
  for (int idx = lane; idx < 16 * FIN; idx += 32) {
    int r = idx / FIN;
    int c = idx - r * FIN;
    float v = 0.f;
    int grow = base + r;
    if (grow < count) {
      int node = lists[d * N_NODES + grow];
      v = aggr[node * FIN + c] * inv_d;
    }
    As[r * 81 + c] = v;
  }
  __syncthreads();

  const int nhalf = lane >> 4;              // 0: K={0,1}/rows 0-7 side, 1: K={2,3}/rows 8-15 side
  const int l15 = lane & 15;
  const int koff = nhalf * 2;

  v8f acc0 = {}, acc1 = {}, acc2 = {}, acc3 = {};
  for (int kk = 0; kk < FIN / 4; ++kk) {
    const int k0 = kk * 4;
    // A fragment: lanes 0-15 hold row l15 K=k0+{0,1}; lanes 16-31 K=k0+{2,3}
    v2f a;
    a.x = As[l15 * 81 + k0 + koff];
    a.y = As[l15 * 81 + k0 + koff + 1];
    // B fragments: lanes = N column, element j = row k0+koff+j
    const float* w0 = W + (k0 + koff) * NV + l15;
    const float* w1 = W + (k0 + koff + 1) * NV + l15;
    v2f b0; b0.x = w0[0];  b0.y = w1[0];
    v2f b1; b1.x = w0[16]; b1.y = w1[16];
    v2f b2; b2.x = w0[32]; b2.y = w1[32];
    v2f b3; b3.x = w0[48]; b3.y = w1[48];
    acc0 = __builtin_amdgcn_wmma_f32_16x16x4_f32(false, a, false, b0, (short)0, acc0, false, false);
    acc1 = __builtin_amdgcn_wmma_f32_16x16x4_f32(false, a, false, b1, (short)0, acc1, false, false);
    acc2 = __builtin_amdgcn_wmma_f32_16x16x4_f32(false, a, false, b2, (short)0, acc2, false, false);
    acc3 = __builtin_amdgcn_wmma_f32_16x16x4_f32(false, a, false, b3, (short)0, acc3, false, false);
  }

  // C/D layout: VGPR i holds row i (lanes 0-15) / row i+8 (lanes 16-31), col = l15 (+16*nt)
  const int rowbase = nhalf * 8;
  int nodes[8];
#pragma unroll
  for (int i = 0; i < 8; ++i) {
    int grow = base + rowbase + i;
    nodes[i] = (grow < count) ? lists[d * N_NODES + grow] : -1;
  }
#pragma unroll
  for (int i = 0; i < 8; ++i) {
    if (nodes[i] < 0) continue;
    float* zr = zout + nodes[i] * NV + l15;
    zr[0]  = 1.f / (1.f + __expf(-acc0[i]));
    zr[16] = 1.f / (1.f + __expf(-acc1[i]));
    zr[32] = 1.f / (1.f + __expf(-acc2[i]));
    zr[48] = 1.f / (1.f + __expf(-acc3[i]));
  }
}

// ---------------- readout: softmax(z @ R) summed per graph ----------------
__global__ __launch_bounds__(256)
void readout(const float* __restrict__ z, const float* __restrict__ R,
             const int* __restrict__ batch, float* __restrict__ out) {
  __shared__ float Rsh[NV * NOUT];          // 640 floats
  __shared__ float accsh[NGRAPH * NOUT];    // 320 floats
  int tid = threadIdx.x;
  for (int i = tid; i < NV * NOUT; i += 256) Rsh[i] = R[i];
  for (int i = tid; i < NGRAPH * NOUT; i += 256) accsh[i] = 0.f;
  __syncthreads();

  int v = blockIdx.x * 256 + tid;
  if (v < N_NODES) {
    float zr[NV];
    const float4* zp = (const float4*)(z + v * NV);
#pragma unroll
    for (int i = 0; i < NV / 4; ++i) {
      float4 t = zp[i];
      zr[4 * i] = t.x; zr[4 * i + 1] = t.y; zr[4 * i + 2] = t.z; zr[4 * i + 3] = t.w;
    }
    float logits[NOUT];
#pragma unroll
    for (int j = 0; j < NOUT; ++j) logits[j] = 0.f;
    for (int k = 0; k < NV; ++k) {
      float zk = zr[k];
#pragma unroll
      for (int j = 0; j < NOUT; ++j) logits[j] += zk * Rsh[k * NOUT + j];
    }
    float mx = logits[0];
#pragma unroll
    for (int j = 1; j < NOUT; ++j) mx = fmaxf(mx, logits[j]);
    float s = 0.f;
#pragma unroll
    for (int j = 0; j < NOUT; ++j) { logits[j] = __expf(logits[j] - mx); s += logits[j]; }
    float inv = 1.f / s;
    int b = batch[v];
#pragma unroll
    for (int j = 0; j < NOUT; ++j) atomicAdd(&accsh[b * NOUT + j], logits[j] * inv);
  }
  __syncthreads();
  for (int i = tid; i < NGRAPH * NOUT; i += 256) atomicAdd(&out[i], accsh[i]);
}

// ---------------- host-side orchestration ----------------
extern "C" void kernel_launch(void* const* d_in, const int* in_sizes, int n_in,
                              void* d_out, int out_size, void* d_ws, size_t ws_size,
                              hipStream_t stream) {
  const float* x     = (const float*)d_in[0];
  const float* ea    = (const float*)d_in[1];
  const float* lw    = (const float*)d_in[2];   // [T, D*FIN*NV]
  const float* rw    = (const float*)d_in[3];   // [T, NV*NOUT]
  const int*   eidx  = (const int*)d_in[4];     // [2, E]
  const int*   deg   = (const int*)d_in[5];
  const int*   batch = (const int*)d_in[6];
  float* out = (float*)d_out;

  char* ws = (char*)d_ws;
  float* aggr = (float*)ws;                                      // N*FIN f32
  float* z0   = (float*)(ws + (size_t)N_NODES * FIN * 4);        // N*NV
  float* z1   = z0 + (size_t)N_NODES * NV;
  int*   counts = (int*)((char*)(z1 + (size_t)N_NODES * NV));
  int*   lists  = (int*)((char*)counts + 256);                   // D*N ints

  const int* src = eidx;
  const int* tgt = eidx + N_EDGES;

  zero_f32<<<1, 256, 0, stream>>>(out, NGRAPH * NOUT);
  zero_f32<<<1, 64, 0, stream>>>((float*)counts, DBUCKETS);      // zero bits == int 0
  build_buckets<<<(N_NODES + 255) / 256, 256, 0, stream>>>(deg, counts, lists);

  const float* zsrc = x;
  float* zdst = z0;
  const long long tot = (long long)N_EDGES * FIN;
  for (int t = 0; t < T_STEPS; ++t) {
    zero_f32<<<4096, 256, 0, stream>>>(aggr, N_NODES * FIN);
    edge_scatter<<<(int)((tot + 255) / 256), 256, 0, stream>>>(zsrc, ea, src, tgt, aggr);
    dim3 grid((N_NODES + 15) / 16, DBUCKETS);
    bucket_linear_wmma<<<grid, 32, 0, stream>>>(
        aggr, lw + (size_t)t * DBUCKETS * FIN * NV, counts, lists, zdst);
    readout<<<(N_NODES + 255) / 256, 256, 0, stream>>>(
        zdst, rw + (size_t)t * NV * NOUT, batch, out);
    zsrc = zdst;
    zdst = (zdst == z0) ? z1 : z0;
  }
}